// HiPPO_85770496901431
// MI455X (gfx1250) — compile-verified
//
#include <hip/hip_runtime.h>
#include <cmath>

typedef __attribute__((ext_vector_type(2))) float v2f;
typedef __attribute__((ext_vector_type(8))) float v8f;

#define SEQ_LEN 512
#define NDIM    256
#define NBATCH  32

// output layout (floats): c_all | y_all | GBT_A | GBT_B
#define OFF_CALL 0
#define OFF_Y    (SEQ_LEN * NBATCH * NDIM)                 //  4,194,304
#define OFF_GA   (OFF_Y + SEQ_LEN * NBATCH)                //  4,210,688
#define OFF_GB   (OFF_GA + (size_t)SEQ_LEN * NDIM * NDIM)  // 37,765,120

#define GA_STRIDE 260   // 256 + 4 pad: rows land on distinct LDS banks
#define CT_STRIDE 260

// ---------------------------------------------------------------------------
// Phase A: discretization. Exploits A = -tril_strict(p p^T) - diag(n+1) and
// ALPHA=0.5 (=> GA = 2*inv(P1) - I). Column j of inv(P1) via O(N) forward
// recurrence with running scalar s = sum p_k x_k. One lane per column;
// the i-loop is uniform so stores to GA[t][i][j0..j0+255] are coalesced.
// ---------------------------------------------------------------------------
__global__ __launch_bounds__(256) void hippo_disc(float* __restrict__ GA,
                                                  float* __restrict__ GB) {
  const int t = blockIdx.x;
  const int j = threadIdx.x;
  const float h  = 1.0f / (float)(t + 1);
  const float h2 = 0.5f * h;

  float* GAt = GA + (size_t)t * NDIM * NDIM;

  const float pj = sqrtf(2.0f * (float)j + 1.0f);
  const float Dj = 1.0f + h2 * (float)(j + 1);

  float x = 0.0f;
  float s = 0.0f;
  for (int i = 0; i < NDIM; ++i) {
    const float pi = sqrtf(2.0f * (float)i + 1.0f);
    const float Di = 1.0f + h2 * (float)(i + 1);
    float val;
    if (i < j) {
      val = 0.0f;                       // GA strictly upper = 0 (exact)
    } else if (i == j) {
      x = 1.0f / Dj;
      s = pj * x;
      val = 2.0f * x - 1.0f;            // diag of 2*inv(P1) - I
    } else {
      x = -(h2 * pi * s) / Di;
      s += pi * x;
      val = 2.0f * x;
    }
    GAt[(size_t)i * NDIM + j] = val;
  }

  if (j == 0) {                         // GB_t = h * inv(P1) @ p  (serial O(N))
    float* GBt = GB + (size_t)t * NDIM;
    float sb = 0.0f;
    for (int i = 0; i < NDIM; ++i) {
      const float pi = sqrtf(2.0f * (float)i + 1.0f);
      const float Di = 1.0f + h2 * (float)(i + 1);
      const float xb = (pi - h2 * pi * sb) / Di;
      sb += pi * xb;
      GBt[i] = h * xb;
    }
  }
}

// ---------------------------------------------------------------------------
// Phase B: sequential scan. One workgroup, 1024 threads = 32 waves.
// Wave w -> output tile (Mtile = w>>1, Ntile = w&1) of the 256x32 state.
// GA_t is lower triangular: stage only the triangle (LDS pre-zeroed once),
// and clamp each tile's K-loop to kmax = 4*(Mtile+1). kmax is identical on
// all 32 lanes of a wave, so readfirstlane pins it to an SGPR => scalar
// loop, EXEC stays all-ones around WMMA (ISA requirement). Prefetch
// GA_{t+1}'s triangle during step t's compute.
// ---------------------------------------------------------------------------
__global__ __launch_bounds__(1024) void hippo_scan(const float* __restrict__ GA,
                                                   const float* __restrict__ GB,
                                                   const float* __restrict__ f,
                                                   float* __restrict__ c_all) {
  extern __shared__ float lds[];
  float* ga = lds;                           // 256 * 260
  float* cT = ga + NDIM * GA_STRIDE;         //  32 * 260  (state, [b][n])
  float* gb = cT + NBATCH * CT_STRIDE;       // 256

  const int tid   = threadIdx.x;
  const int wave  = tid >> 5;
  const int lane  = tid & 31;
  const int lo    = lane & 15;
  const int hi    = lane >> 4;               // selects K pair / M half
  const int Mtile = wave >> 1;               // 0..15
  const int Ntile = wave & 1;                // 0..1
  const int bcol  = Ntile * 16 + lo;         // batch index (B-matrix column)
  const int arow  = Mtile * 16 + lo;         // A-matrix row
  // triangular K bound; wave-uniform -> pin to SGPR for a scalar loop
  const int kmax  = __builtin_amdgcn_readfirstlane(4 * (Mtile + 1));

  // one-time zero: ga (upper triangle + pad stays 0 forever) and state c0
  for (int idx = tid; idx < NDIM * GA_STRIDE + NBATCH * CT_STRIDE; idx += 1024)
    ga[idx] = 0.0f;

  const float* arow_base = ga + (size_t)arow * GA_STRIDE + 2 * hi;
  const float* bcol_base = cT + (size_t)bcol * CT_STRIDE + 2 * hi;

  for (int t = 0; t < SEQ_LEN; ++t) {
    __syncthreads();  // prior step done reading ga/cT

    // stage lower triangle of GA_t; chunk (row, c4) needed iff 4*c4 <= row
    const float4* src = (const float4*)(GA + (size_t)t * NDIM * NDIM);
    for (int ch = tid; ch < NDIM * 64; ch += 1024) {
      const int row = ch >> 6;
      const int c4  = ch & 63;
      if ((c4 << 2) <= row) {
        const float4 v = src[row * 64 + c4];
        *(float4*)(ga + (size_t)row * GA_STRIDE + c4 * 4) = v;  // 16B aligned (pitch 1040B)
      }
    }
    if (tid < NDIM) gb[tid] = GB[(size_t)t * NDIM + tid];

    // prefetch next step's triangle (fire-and-forget -> global_prefetch_b8)
    if (t + 1 < SEQ_LEN) {
      const float* nxt = GA + (size_t)(t + 1) * NDIM * NDIM;
      const int prow = tid >> 2;             // each thread: 64 floats = 2 lines
      const int pcol = (tid & 3) * 64;
      if (pcol <= prow) {
        __builtin_prefetch(nxt + (size_t)prow * NDIM + pcol, 0, 0);
        __builtin_prefetch(nxt + (size_t)prow * NDIM + pcol + 32, 0, 0);
      }
    }
    __syncthreads();

    v8f acc = {0.f, 0.f, 0.f, 0.f, 0.f, 0.f, 0.f, 0.f};
#pragma unroll 4
    for (int k = 0; k < kmax; ++k) {
      const v2f a = *(const v2f*)(arow_base + 4 * k);   // GA[arow][4k+2hi .. +1]
      const v2f b = *(const v2f*)(bcol_base + 4 * k);   // c_old[4k+2hi .. +1][bcol]
      acc = __builtin_amdgcn_wmma_f32_16x16x4_f32(
          /*neg_a=*/false, a, /*neg_b=*/false, b,
          /*c_mod=*/(short)0, acc, /*reuse_a=*/false, /*reuse_b=*/false);
    }

    // + GB_t[M] * f[b][t]
    const float fb = f[(size_t)bcol * SEQ_LEN + t];
#pragma unroll
    for (int r = 0; r < 8; ++r) {
      acc[r] += gb[Mtile * 16 + 8 * hi + r] * fb;
    }

    __syncthreads();  // all waves finished reading c_old before overwrite

    // commit: LDS state (transposed) + c_all[t][b][n] (8 consecutive n per lane)
    const int mbase = Mtile * 16 + 8 * hi;
#pragma unroll
    for (int r = 0; r < 8; ++r) {
      cT[(size_t)bcol * CT_STRIDE + mbase + r] = acc[r];
    }
    float* outp = c_all + (size_t)t * NBATCH * NDIM + (size_t)bcol * NDIM + mbase;
    float4 o0 = {acc[0], acc[1], acc[2], acc[3]};
    float4 o1 = {acc[4], acc[5], acc[6], acc[7]};
    *(float4*)(outp)     = o0;
    *(float4*)(outp + 4) = o1;
  }
}

// ---------------------------------------------------------------------------
// Phase C: y[t][b] = sum_n C[n] * c_all[t][b][n]  (parallel over t,b)
// ---------------------------------------------------------------------------
__global__ __launch_bounds__(256) void hippo_y(const float* __restrict__ c_all,
                                               const float* __restrict__ Cvec,
                                               float* __restrict__ y_all) {
  __shared__ float red[256];
  const int tb = blockIdx.x;        // t*32 + b
  const int n  = threadIdx.x;
  red[n] = c_all[(size_t)tb * NDIM + n] * Cvec[n];
  __syncthreads();
  for (int off = 128; off > 0; off >>= 1) {
    if (n < off) red[n] += red[n + off];
    __syncthreads();
  }
  if (n == 0) y_all[tb] = red[0];
}

extern "C" void kernel_launch(void* const* d_in, const int* in_sizes, int n_in,
                              void* d_out, int out_size, void* d_ws, size_t ws_size,
                              hipStream_t stream) {
  const float* f    = (const float*)d_in[0];  // (32, 512, 1)
  const float* Cvec = (const float*)d_in[3];  // (256,)
  float* out   = (float*)d_out;
  float* c_all = out + OFF_CALL;
  float* y_all = out + OFF_Y;
  float* GA    = out + OFF_GA;
  float* GB    = out + OFF_GB;

  hippo_disc<<<SEQ_LEN, 256, 0, stream>>>(GA, GB);

  const size_t lds_bytes =
      (size_t)(NDIM * GA_STRIDE + NBATCH * CT_STRIDE + NDIM) * sizeof(float);  // 300,544 B
  hippo_scan<<<1, 1024, lds_bytes, stream>>>(GA, GB, f, c_all);

  hippo_y<<<SEQ_LEN * NBATCH, 256, 0, stream>>>(c_all, Cvec, y_all);
}